// ProjectNet_36807869727342
// MI455X (gfx1250) — compile-verified
//
#include <hip/hip_runtime.h>

// ---------------------------------------------------------------------------
// ProjectNet on MI455X (gfx1250): fp32 WMMA (V_WMMA_F32_16X16X4_F32) for the
// learned-step GEMM (512x65536x256) and the Dykstra projection GEMM chain
// (60 iters x [512,512]@[512,256] + [512,256]@[256,512] per round).
// ~100 GFLOP fp32 total; MM_w (67MB) is L2-resident (192MB L2) -> compute
// bound on the matrix pipes, so everything runs through WMMA tiles.
// ---------------------------------------------------------------------------

typedef float v2f __attribute__((ext_vector_type(2)));
typedef float v8f __attribute__((ext_vector_type(8)));

#define DIM    256
#define NFULL  512
#define MCON   256
#define BATCH  512
#define ROUNDS 3
#define PITERS 60
#define TOLF   1e-3f
#define RHO    0.1f
#define XRHO   0.1f
#define GU     0.5f
#define GO     0.5f

__device__ __forceinline__ v8f wmma_f32(v2f a, v2f b, v8f c) {
  // D = A(16x4 f32) x B(4x16 f32) + C(16x16 f32)
  return __builtin_amdgcn_wmma_f32_16x16x4_f32(
      /*neg_a=*/false, a, /*neg_b=*/false, b,
      /*c_mod=*/(short)0, c, /*reuse_a=*/false, /*reuse_b=*/false);
}

// ---------------------------------------------------------------------------
// zero-fill
__global__ void k_zero(float* __restrict__ p, int n) {
  int i = blockIdx.x * blockDim.x + threadIdx.x;
  if (i < n) p[i] = 0.f;
}

// ---------------------------------------------------------------------------
// Fused: M[b,i,j] = sum_k diff[b,k]*W[i*256+j,k] + Mb[i*256+j]
//        s[b,i]   = sum_j M[b,i,j]*x[b,j]
// One wave per (b-tile, i). 8 waves/block share LDS-staged diff & x tiles.
// grid = (BATCH/16, DIM/8), block = 256.
__global__ __launch_bounds__(256)
void k_step_s(const float* __restrict__ xc, const float* __restrict__ c,
              const float* __restrict__ W,  const float* __restrict__ Mb,
              float* __restrict__ S) {
  __shared__ float diffL[16][258];  // +2 pad: avoid 64-bank conflicts, keep 8B align
  __shared__ float xL[16][258];
  const int b0 = blockIdx.x * 16;
  const int i  = blockIdx.y * 8 + (threadIdx.x >> 5);

  for (int idx = threadIdx.x; idx < 16 * DIM; idx += 256) {
    int r = idx >> 8, col = idx & 255;
    float xv = xc[(b0 + r) * DIM + col];
    xL[r][col]    = xv;
    diffL[r][col] = xv - c[(b0 + r) * DIM + col];
  }
  __syncthreads();

  const int l = threadIdx.x & 31, hi = l >> 4, lo = l & 15;
  float sacc[8];
#pragma unroll
  for (int v = 0; v < 8; ++v) sacc[v] = 0.f;

  for (int jt = 0; jt < 16; ++jt) {
    const int j0 = jt * 16;
    v8f acc = {0.f, 0.f, 0.f, 0.f, 0.f, 0.f, 0.f, 0.f};
    const float* wrow = W + (size_t)(i * DIM + j0 + lo) * DIM;  // B[k,n]=W[u0+n,k]
    for (int k0 = 0; k0 < DIM; k0 += 4) {
      const int ks = k0 + 2 * hi;
      v2f a;  a.x  = diffL[lo][ks]; a.y  = diffL[lo][ks + 1];   // A[m=lo, k]
      v2f bb; bb.x = wrow[ks];      bb.y = wrow[ks + 1];
      acc = wmma_f32(a, bb, acc);
    }
    const float mb = Mb[i * DIM + j0 + lo];
#pragma unroll
    for (int v = 0; v < 8; ++v) {
      const int bl = v + 8 * hi;                    // C row = b-local
      sacc[v] += (acc[v] + mb) * xL[bl][j0 + lo];   // * x[b, j], partial sum over j
    }
  }
  // reduce over the 16 lanes of each half-wave (the 16 j columns)
#pragma unroll
  for (int v = 0; v < 8; ++v) {
    float t = sacc[v];
    t += __shfl_xor(t, 1, 32);
    t += __shfl_xor(t, 2, 32);
    t += __shfl_xor(t, 4, 32);
    t += __shfl_xor(t, 8, 32);
    sacc[v] = t;
  }
  if (lo == 0) {
#pragma unroll
    for (int v = 0; v < 8; ++v) {
      const int b = b0 + v + 8 * hi;
      S[b * DIM + i] = sacc[v];
    }
  }
}

// ---------------------------------------------------------------------------
// x0 = x - (xrho*s + rho*grad + rho*diff); init Dykstra state (XP=x0 pad, Q=0,
// YOUT=x0 pad) and reset done/err flags.
__global__ void k_round_update(const float* __restrict__ xc, const float* __restrict__ c,
                               const float* __restrict__ S,
                               float* __restrict__ XP, float* __restrict__ Q,
                               float* __restrict__ YOUT, int* __restrict__ flags) {
  const int idx = blockIdx.x * blockDim.x + threadIdx.x;
  if (idx >= BATCH * NFULL) return;
  const int b = idx / NFULL, n = idx % NFULL;
  float x0 = 0.f;
  if (n < DIM) {
    const float x = xc[b * DIM + n], cv = c[b * DIM + n];
    const float diff = x - cv;
    const float grad = (cv >= x ? -GU : 0.f) + (x >= cv ? GO : 0.f);
    x0 = x - (XRHO * S[b * DIM + n] + RHO * grad + RHO * diff);
  }
  XP[idx] = x0; Q[idx] = 0.f; YOUT[idx] = x0;
  if (idx == 0) { flags[0] = 0; flags[1] = 0; }
}

// ---------------------------------------------------------------------------
// T[b,m] = sum_k XP[b,k]*A[m,k] - bvec[m].   One wave per 16x16 tile.
// grid = 32(b-tiles)*16(m-tiles)/4 blocks of 128 threads.
__global__ __launch_bounds__(128)
void k_proj_t(const float* __restrict__ XP, const float* __restrict__ Amat,
              const float* __restrict__ bvec, float* __restrict__ T,
              const int* __restrict__ flags) {
  if (flags[0]) return;  // frozen (uniform branch; EXEC intact otherwise)
  const int w  = blockIdx.x * 4 + (threadIdx.x >> 5);
  const int b0 = (w >> 4) * 16;
  const int m0 = (w & 15) * 16;
  const int l = threadIdx.x & 31, hi = l >> 4, lo = l & 15;
  v8f acc = {0.f, 0.f, 0.f, 0.f, 0.f, 0.f, 0.f, 0.f};
  const float* xprow = XP   + (size_t)(b0 + lo) * NFULL;
  const float* arow  = Amat + (size_t)(m0 + lo) * NFULL;
  for (int k0 = 0; k0 < NFULL; k0 += 4) {
    const int ks = k0 + 2 * hi;
    v2f a;  a.x  = xprow[ks]; a.y  = xprow[ks + 1];
    v2f bb; bb.x = arow[ks];  bb.y = arow[ks + 1];
    acc = wmma_f32(a, bb, acc);
  }
  const float bv = bvec[m0 + lo];
#pragma unroll
  for (int v = 0; v < 8; ++v) {
    const int b = b0 + v + 8 * hi;
    T[b * MCON + m0 + lo] = acc[v] - bv;
  }
}

// ---------------------------------------------------------------------------
// Y = XP - T @ AA.T ; then Dykstra elementwise updates + global err max.
// grid = 32(b)*32(n)/4 blocks of 128 threads.
__global__ __launch_bounds__(128)
void k_proj_y(const float* __restrict__ T, const float* __restrict__ AA,
              float* __restrict__ XP, float* __restrict__ Q,
              float* __restrict__ YOUT, int* __restrict__ flags) {
  if (flags[0]) return;
  const int w  = blockIdx.x * 4 + (threadIdx.x >> 5);
  const int b0 = (w >> 5) * 16;
  const int n0 = (w & 31) * 16;
  const int l = threadIdx.x & 31, hi = l >> 4, lo = l & 15;
  v8f acc = {0.f, 0.f, 0.f, 0.f, 0.f, 0.f, 0.f, 0.f};
  const float* trow  = T  + (size_t)(b0 + lo) * MCON;
  const float* aarow = AA + (size_t)(n0 + lo) * MCON;  // AA.T[m,n] = AA[n,m]
  for (int m0 = 0; m0 < MCON; m0 += 4) {
    const int ms = m0 + 2 * hi;
    v2f a;  a.x  = trow[ms];  a.y  = trow[ms + 1];
    v2f bb; bb.x = aarow[ms]; bb.y = aarow[ms + 1];
    acc = wmma_f32(a, bb, acc);
  }
  float lmax = 0.f;
#pragma unroll
  for (int v = 0; v < 8; ++v) {
    const int b = b0 + v + 8 * hi;
    const int n = n0 + lo;
    const size_t ix = (size_t)b * NFULL + n;
    const float xp = XP[ix];
    const float y  = xp - acc[v];
    const float q  = Q[ix];
    const float yq = y + q;
    const float xn = fmaxf(yq, 0.f);   // x_new = relu(y+q)
    const float pn = xp - y;           // p_new = (x+p) - y
    XP[ix]   = xn + pn;                // next iterate of x+p
    Q[ix]    = yq - xn;
    YOUT[ix] = y;
    lmax = fmaxf(lmax, fmaxf(-y, 0.f));
  }
  lmax = fmaxf(lmax, __shfl_xor(lmax, 1, 32));
  lmax = fmaxf(lmax, __shfl_xor(lmax, 2, 32));
  lmax = fmaxf(lmax, __shfl_xor(lmax, 4, 32));
  lmax = fmaxf(lmax, __shfl_xor(lmax, 8, 32));
  lmax = fmaxf(lmax, __shfl_xor(lmax, 16, 32));
  if (l == 0)
    atomicMax((unsigned int*)&flags[1], __float_as_uint(lmax));  // lmax>=0: bit-monotone
}

__global__ void k_proj_check(int* __restrict__ flags) {
  if (blockIdx.x == 0 && threadIdx.x == 0) {
    if (!flags[0] && __uint_as_float(((unsigned int*)flags)[1]) < TOLF) flags[0] = 1;
    flags[1] = 0;
  }
}

__global__ void k_extract(const float* __restrict__ YOUT, float* __restrict__ dst) {
  const int idx = blockIdx.x * blockDim.x + threadIdx.x;
  if (idx >= BATCH * DIM) return;
  const int b = idx >> 8, i = idx & 255;
  dst[idx] = YOUT[b * NFULL + i];
}

// ---------------------------------------------------------------------------
extern "C" void kernel_launch(void* const* d_in, const int* in_sizes, int n_in,
                              void* d_out, int out_size, void* d_ws, size_t ws_size,
                              hipStream_t stream) {
  const float* c   = (const float*)d_in[0];  // [512,256]
  const float* A   = (const float*)d_in[1];  // [256,512]
  const float* bv  = (const float*)d_in[2];  // [256]
  const float* AA  = (const float*)d_in[3];  // [512,256]
  const float* W   = (const float*)d_in[4];  // [65536,256]
  const float* Mb  = (const float*)d_in[5];  // [65536]

  float* ws = (float*)d_ws;
  float* XP = ws;                       // [512,512] x+p
  float* Q  = XP + BATCH * NFULL;       // [512,512]
  float* YO = Q  + BATCH * NFULL;       // [512,512] y_out
  float* T  = YO + BATCH * NFULL;       // [512,256]
  float* S  = T  + BATCH * MCON;        // [512,256]
  float* XC = S  + BATCH * DIM;         // [512,256] current x
  int* flags = (int*)(XC + BATCH * DIM);  // [done, err_bits]

  k_zero<<<(BATCH * DIM + 255) / 256, 256, 0, stream>>>(XC, BATCH * DIM);

  for (int r = 0; r < ROUNDS; ++r) {
    dim3 gs(BATCH / 16, DIM / 8);
    k_step_s<<<gs, 256, 0, stream>>>(XC, c, W, Mb, S);
    k_round_update<<<(BATCH * NFULL + 255) / 256, 256, 0, stream>>>(XC, c, S, XP, Q, YO, flags);
    for (int it = 0; it < PITERS; ++it) {
      k_proj_t<<<(BATCH / 16) * (MCON / 16) / 4, 128, 0, stream>>>(XP, A, bv, T, flags);
      k_proj_y<<<(BATCH / 16) * (NFULL / 16) / 4, 128, 0, stream>>>(T, AA, XP, Q, YO, flags);
      k_proj_check<<<1, 1, 0, stream>>>(flags);
    }
    k_extract<<<(BATCH * DIM + 255) / 256, 256, 0, stream>>>(
        YO, (r == ROUNDS - 1) ? (float*)d_out : XC);
  }
}